// GraphUnpooling_19061064859667
// MI455X (gfx1250) — compile-verified
//
#include <hip/hip_runtime.h>

// GraphUnpooling gather: out[b, nf, :, :] = x[b, f2c[nf], :, :]
// B=4, NC=2000, F=4, H=64, NF=50000 (F,H,B fixed by the reference; NC/NF
// derived from in_sizes). Chunk per (b,node) = F*H = 256 floats = 64 float4.
//
// Memory-bound: 204.8 MB written (> 192 MB L2), source x (8.2 MB) is reused
// ~25x and must stay L2-resident. Stores are marked non-temporal (TH=NT) so
// the output stream doesn't evict x; loads stay RT. HBM floor ~9 us.

#define B_DIM  4
#define F_DIM  4
#define H_DIM  64
#define CHUNK_V4 ((F_DIM * H_DIM) / 4)   // 64 float4 per (b, node)

// gfx12+ CPol immediate: TH in bits [2:0], SCOPE in bits [4:3]
#define CPOL_RT  0   // regular temporal, WGP scope
#define CPOL_NT  1   // non-temporal,    WGP scope

// Native 16-byte vector types (builtins reject HIP_vector_type classes)
typedef float nfloat4 __attribute__((vector_size(4 * sizeof(float))));

// ---------------------------------------------------------------------------
// gfx1250 async global<->LDS path (ASYNCcnt-tracked), guarded by __has_builtin
// ---------------------------------------------------------------------------
#if defined(__has_builtin)
#if __has_builtin(__builtin_amdgcn_global_load_async_to_lds_b128) && \
    __has_builtin(__builtin_amdgcn_global_store_async_from_lds_b128) && \
    __has_builtin(__builtin_amdgcn_s_wait_asynccnt)
#define HAVE_ASYNC_LDS 1
#endif
#endif

#ifndef HAVE_ASYNC_LDS
#define HAVE_ASYNC_LDS 0
#endif

#if HAVE_ASYNC_LDS
// Parameter types from clang's diagnostics:
//   arg0: '__attribute__((__vector_size__(4 * sizeof(int)))) int *'            (generic)
//   arg1: '__attribute__((__vector_size__(4 * sizeof(int)))) int __shared__ *' (AS3)
typedef int v4i __attribute__((vector_size(4 * sizeof(int))));
typedef __attribute__((address_space(3))) v4i as3_v4i;

__global__ __launch_bounds__(256) void gather_unpool_async(
    const float4* __restrict__ x,    // [B, NC, 64] as float4
    const int*    __restrict__ f2c,  // [NF]
    float4*       __restrict__ out,  // [B, NF, 64] as float4
    int NC, int NF)
{
  __shared__ float4 stage[256];      // 4 KB staging, one float4 slot per lane

  const int t  = blockIdx.x * 256 + threadIdx.x;  // [0, NF*64)
  const int b  = blockIdx.y;                      // [0, B)
  const int nf = t >> 6;                          // fine node
  const int r  = t & 63;                          // float4 within chunk
  if (nf >= NF) return;

  const int c = f2c[nf];                          // parent coarse node

  const float4* src = x   + ((size_t)(b * NC + c)  * CHUNK_V4 + r);
  float4*       dst = out + ((size_t)(b * NF + nf) * CHUNK_V4 + r);

  as3_v4i* lds = (as3_v4i*)&stage[threadIdx.x];

  // global -> LDS (16B/lane), RT so x allocates/stays in L2; ASYNCcnt-tracked
  __builtin_amdgcn_global_load_async_to_lds_b128((v4i*)src, lds, 0, CPOL_RT);
  // own-wave load must have landed in LDS before the store reads it
  __builtin_amdgcn_s_wait_asynccnt(0);
  // LDS -> global (16B/lane), NT so the 205 MB output stream doesn't evict x
  __builtin_amdgcn_global_store_async_from_lds_b128((v4i*)dst, lds, 0, CPOL_NT);
  // drain before LDS deallocation / endpgm (endpgm also waits idle)
  __builtin_amdgcn_s_wait_asynccnt(0);
}
#endif

// ---------------------------------------------------------------------------
// Fallback: direct VGPR float4 copy (global_load_b128 / global_store_b128 nt)
// ---------------------------------------------------------------------------
__global__ __launch_bounds__(256) void gather_unpool_vec4(
    const float4* __restrict__ x,
    const int*    __restrict__ f2c,
    float4*       __restrict__ out,
    int NC, int NF)
{
  const int t  = blockIdx.x * 256 + threadIdx.x;
  const int b  = blockIdx.y;
  const int nf = t >> 6;
  const int r  = t & 63;
  if (nf >= NF) return;

  const int c = f2c[nf];
  const nfloat4* srcv = (const nfloat4*)(x   + (size_t)(b * NC + c)  * CHUNK_V4 + r);
  nfloat4*       dstv = (nfloat4*)      (out + (size_t)(b * NF + nf) * CHUNK_V4 + r);
  const nfloat4 v = *srcv;
  __builtin_nontemporal_store(v, dstv);
}

extern "C" void kernel_launch(void* const* d_in, const int* in_sizes, int n_in,
                              void* d_out, int out_size, void* d_ws, size_t ws_size,
                              hipStream_t stream) {
  const float4* x   = (const float4*)d_in[0];  // [B, NC, F, H] float32
  const int*    f2c = (const int*)d_in[1];     // [NF] int32
  float4*       out = (float4*)d_out;          // [B, NF, F, H] float32

  const int NF = in_sizes[1];
  const int NC = in_sizes[0] / (B_DIM * F_DIM * H_DIM);

  const int total_v4 = NF * CHUNK_V4;          // float4 slots per batch
  dim3 grid((total_v4 + 255) / 256, B_DIM, 1);
  dim3 block(256, 1, 1);

#if HAVE_ASYNC_LDS
  gather_unpool_async<<<grid, block, 0, stream>>>(x, f2c, out, NC, NF);
#else
  gather_unpool_vec4<<<grid, block, 0, stream>>>(x, f2c, out, NC, NF);
#endif
}